// RecGRU_W_up_42691974922286
// MI455X (gfx1250) — compile-verified
//
#include <hip/hip_runtime.h>

// GConvGRU (Chebyshev K=1-style rescaled-Laplacian GRU) for MI455X / gfx1250.
// N=24 nodes, C=4096 channels, E=96 edges. Memory-bound on 12 x 64MB fp32
// weight matrices (~805 MB > 192 MB L2) -> fp32 WMMA (16x16x4) keeps full
// reference precision and exceeds the ~280 TF/s needed to saturate 23.3 TB/s
// HBM on this skinny (M=24) GEMM shape. Weights are streamed with NT hints;
// the tiny reused activations stay resident in L2.

#define NODES 24
#define CH    4096
#define NEDGE 96

typedef __attribute__((ext_vector_type(2))) float v2f;
typedef __attribute__((ext_vector_type(8))) float v8f;

__device__ __forceinline__ v8f wmma_f32(v2f a, v2f b, v8f c) {
  // D = A(16x4) * B(4x16) + C(16x16), fp32 in/out.
  return __builtin_amdgcn_wmma_f32_16x16x4_f32(
      /*neg_a=*/false, a, /*neg_b=*/false, b,
      /*c_mod=*/(short)0, c, /*reuse_a=*/false, /*reuse_b=*/false);
}

// ---------------------------------------------------------------- zero fill
__global__ void zero_kernel(float* __restrict__ p, int n) {
  const int i = blockIdx.x * blockDim.x + threadIdx.x;
  if (i < n) p[i] = 0.f;
}

// ---------------------------------------------------------------- norm[e]
__global__ void norm_kernel(const int* __restrict__ ei,
                            const float* __restrict__ ew,
                            float* __restrict__ nrm) {
  __shared__ float degs[NODES];
  __shared__ float degd[NODES];
  const int t = threadIdx.x;
  if (t < NODES) { degs[t] = 0.f; degd[t] = 0.f; }
  __syncthreads();
  if (t < NEDGE) {
    atomicAdd(&degs[ei[t]], ew[t]);          // segment_sum over src
    atomicAdd(&degd[ei[NEDGE + t]], ew[t]);  // segment_sum over dst
  }
  __syncthreads();
  if (t < NODES) {
    float ds = degs[t], dd = degd[t];
    degs[t] = (ds > 0.f) ? rsqrtf(fmaxf(ds, 1e-12f)) : 0.f;
    degd[t] = (dd > 0.f) ? rsqrtf(fmaxf(dd, 1e-12f)) : 0.f;
  }
  __syncthreads();
  if (t < NEDGE) nrm[t] = ew[t] * degs[ei[t]] * degd[ei[NEDGE + t]];
}

// ------------------------------------------- LX = (2/lam-1)*T - (2/lam)*AX
__global__ void prop_kernel(const float* __restrict__ T,
                            float* __restrict__ LX,
                            const float* __restrict__ nrm,
                            const int* __restrict__ ei,
                            const float* __restrict__ lambda_max) {
  __shared__ int   s_src[NEDGE];
  __shared__ int   s_dst[NEDGE];
  __shared__ float s_nrm[NEDGE];
  const int t = threadIdx.x;
  if (t < NEDGE) {
    s_src[t] = ei[t];
    s_dst[t] = ei[NEDGE + t];
    s_nrm[t] = nrm[t];
  }
  __syncthreads();
  const int node = blockIdx.y;
  const int c    = blockIdx.x * blockDim.x + t;
  const float s  = 2.0f / lambda_max[0];
  float ax = 0.f;
  #pragma unroll 8
  for (int e = 0; e < NEDGE; ++e) {
    if (s_dst[e] == node)
      ax += s_nrm[e] * T[(size_t)s_src[e] * CH + c];
  }
  const float tv = T[(size_t)node * CH + c];
  LX[(size_t)node * CH + c] = (s - 1.0f) * tv - s * ax;
}

// ------------------------------------------- Y = relu(A[24xK] @ B[KxN])
// One wave per 16-col N-tile; two M-tiles (rows 0..15, 16..23; rows 24..31
// read a zeroed scratch row so there is NO masking in the hot loop) share
// each B fragment. 4 waves / block, 64 blocks.
__global__ __launch_bounds__(128) void gemm_relu_kernel(
    const float* __restrict__ A, const float* __restrict__ B,
    const float* __restrict__ zrow, float* __restrict__ Y) {
  const int wv   = threadIdx.x >> 5;
  const int lane = threadIdx.x & 31;
  const int n0   = ((blockIdx.x << 2) + wv) << 4;  // N-tile start col
  const int h    = lane >> 4;                      // lane-half (K pairing)
  const int mn   = lane & 15;                      // row of A / col of B

  const float* A0 = A + (size_t)mn * CH;                        // rows 0..15
  const bool  r1  = (mn + 16) < NODES;                          // rows 16..23
  const float* A1 = r1 ? (A + (size_t)(mn + 16) * CH) : zrow;   // pad -> zeros

  v8f acc0 = {}; v8f acc1 = {};
  #pragma unroll 8
  for (int k = 0; k < CH; k += 4) {
    if (((k & 255) == 0) && (k + 512 < CH))
      __builtin_prefetch(B + (size_t)(k + 512) * CH + n0 + mn, 0, 1);
    const int kk = k + 2 * h;
    const v2f a0 = *(const v2f*)(A0 + kk);  // contiguous K-pair
    const v2f a1 = *(const v2f*)(A1 + kk);
    v2f b;                                   // B row-major [K,N], read-once:
    b.x = __builtin_nontemporal_load(B + (size_t)kk * CH + n0 + mn);
    b.y = __builtin_nontemporal_load(B + (size_t)(kk + 1) * CH + n0 + mn);
    acc0 = wmma_f32(a0, b, acc0);
    acc1 = wmma_f32(a1, b, acc1);
  }
  #pragma unroll
  for (int v = 0; v < 8; ++v) {
    const int m0 = v + 8 * h;               // C/D layout: VGPR v -> row v+8h
    Y[(size_t)m0 * CH + n0 + mn] = fmaxf(acc0[v], 0.f);
    const int m1 = m0 + 16;
    if (m1 < NODES)
      Y[(size_t)m1 * CH + n0 + mn] = fmaxf(acc1[v], 0.f);
  }
}

// ---------------- Y = act(A1 @ B1^T + A2 @ B2^T + bias) ; gate weights are
// [Nout, K] row-major so B^T fragments are contiguous float2 along K.
// MODE 0: sigmoid -> Y.  MODE 1: tanh, then out = Z*Ht + (1-Z)*H -> Y.
template <int MODE>
__global__ __launch_bounds__(128) void gemm_gate_kernel(
    const float* __restrict__ A1, const float* __restrict__ B1,
    const float* __restrict__ A2, const float* __restrict__ B2,
    const float* __restrict__ bias, const float* __restrict__ Zb,
    const float* __restrict__ Hin, const float* __restrict__ zrow,
    float* __restrict__ Y) {
  const int wv   = threadIdx.x >> 5;
  const int lane = threadIdx.x & 31;
  const int n0   = ((blockIdx.x << 2) + wv) << 4;
  const int h    = lane >> 4;
  const int mn   = lane & 15;

  const float* A1r0 = A1 + (size_t)mn * CH;
  const float* A2r0 = A2 + (size_t)mn * CH;
  const bool  r1    = (mn + 16) < NODES;
  const float* A1r1 = r1 ? (A1 + (size_t)(mn + 16) * CH) : zrow;
  const float* A2r1 = r1 ? (A2 + (size_t)(mn + 16) * CH) : zrow;
  const float* B1r  = B1 + (size_t)(n0 + mn) * CH;  // column n of B1^T
  const float* B2r  = B2 + (size_t)(n0 + mn) * CH;

  v8f acc0 = {}; v8f acc1 = {};
  #pragma unroll 8
  for (int k = 0; k < CH; k += 4) {
    if (((k & 255) == 0) && (k + 512 < CH)) {
      __builtin_prefetch(B1r + k + 512, 0, 1);
      __builtin_prefetch(B2r + k + 512, 0, 1);
    }
    const int kk = k + 2 * h;
    v2f a = *(const v2f*)(A1r0 + kk);
    v2f b = __builtin_nontemporal_load((const v2f*)(B1r + kk));
    acc0 = wmma_f32(a, b, acc0);
    v2f a1 = *(const v2f*)(A1r1 + kk);
    acc1 = wmma_f32(a1, b, acc1);

    a = *(const v2f*)(A2r0 + kk);
    b = __builtin_nontemporal_load((const v2f*)(B2r + kk));
    acc0 = wmma_f32(a, b, acc0);
    a1 = *(const v2f*)(A2r1 + kk);
    acc1 = wmma_f32(a1, b, acc1);
  }
  #pragma unroll
  for (int v = 0; v < 8; ++v) {
    const int m0 = v + 8 * h;
    {
      const size_t i = (size_t)m0 * CH + n0 + mn;
      const float pre = acc0[v] + bias[i];
      if (MODE == 0) {
        Y[i] = 1.0f / (1.0f + expf(-pre));
      } else {
        const float ht = tanhf(pre);
        const float z  = Zb[i];
        Y[i] = z * ht + (1.0f - z) * Hin[i];
      }
    }
    const int m1 = m0 + 16;
    if (m1 < NODES) {
      const size_t i = (size_t)m1 * CH + n0 + mn;
      const float pre = acc1[v] + bias[i];
      if (MODE == 0) {
        Y[i] = 1.0f / (1.0f + expf(-pre));
      } else {
        const float ht = tanhf(pre);
        const float z  = Zb[i];
        Y[i] = z * ht + (1.0f - z) * Hin[i];
      }
    }
  }
}

// ---------------------------------------------------------- HR = H * R
__global__ void mul_kernel(const float* __restrict__ a,
                           const float* __restrict__ b,
                           float* __restrict__ y, int n) {
  const int i = blockIdx.x * blockDim.x + threadIdx.x;
  if (i < n) y[i] = a[i] * b[i];
}

extern "C" void kernel_launch(void* const* d_in, const int* in_sizes, int n_in,
                              void* d_out, int out_size, void* d_ws, size_t ws_size,
                              hipStream_t stream) {
  const float* X    = (const float*)d_in[0];
  const int*   ei   = (const int*)d_in[1];
  const float* ew   = (const float*)d_in[2];
  const float* H    = (const float*)d_in[3];
  const float* lam  = (const float*)d_in[4];
  const float* Wcxz = (const float*)d_in[5];
  const float* Wchz = (const float*)d_in[6];
  const float* Wcxr = (const float*)d_in[7];
  const float* Wchr = (const float*)d_in[8];
  const float* Wcxh = (const float*)d_in[9];
  const float* Wchh = (const float*)d_in[10];
  const float* wxz  = (const float*)d_in[11];
  const float* wqz  = (const float*)d_in[12];
  const float* wxr  = (const float*)d_in[13];
  const float* wqr  = (const float*)d_in[14];
  const float* wxh  = (const float*)d_in[15];
  const float* wqh  = (const float*)d_in[16];
  const float* bz   = (const float*)d_in[17];
  const float* br   = (const float*)d_in[18];
  const float* bh   = (const float*)d_in[19];
  float* out = (float*)d_out;

  // Workspace carve (~4.74 MB of fp32), fully overwritten every call.
  const size_t NC = (size_t)NODES * CH;  // 98304
  float* ws    = (float*)d_ws;
  float* nrm   = ws;           // 96 (pad to 256)
  float* LX_X  = ws + 256;
  float* LX_H  = LX_X + NC;
  float* LX_HR = LX_H + NC;
  float* Cxz   = LX_HR + NC;
  float* Chz   = Cxz + NC;
  float* Cxr   = Chz + NC;
  float* Chr   = Cxr + NC;
  float* Cxh   = Chr + NC;
  float* Chh   = Cxh + NC;
  float* Zbuf  = Chh + NC;
  float* Rbuf  = Zbuf + NC;
  float* HR    = Rbuf + NC;
  float* zrow  = HR + NC;      // one zeroed row: padding for M-tile 1

  const dim3 propGrid(CH / 256, NODES);
  const int  gemmBlocks = (CH / 16) / 4;  // 256 N-tiles, 4 waves/block = 64

  // 0) zero pad-row used by all GEMMs
  zero_kernel<<<CH / 256, 256, 0, stream>>>(zrow, CH);
  // 1) edge normalization
  norm_kernel<<<1, 128, 0, stream>>>(ei, ew, nrm);
  // 2) propagate X and H
  prop_kernel<<<propGrid, 256, 0, stream>>>(X, LX_X, nrm, ei, lam);
  prop_kernel<<<propGrid, 256, 0, stream>>>(H, LX_H, nrm, ei, lam);
  // 3) conv GEMMs (relu(LX @ Wc))
  gemm_relu_kernel<<<gemmBlocks, 128, 0, stream>>>(LX_X, Wcxz, zrow, Cxz);
  gemm_relu_kernel<<<gemmBlocks, 128, 0, stream>>>(LX_H, Wchz, zrow, Chz);
  gemm_relu_kernel<<<gemmBlocks, 128, 0, stream>>>(LX_X, Wcxr, zrow, Cxr);
  gemm_relu_kernel<<<gemmBlocks, 128, 0, stream>>>(LX_H, Wchr, zrow, Chr);
  gemm_relu_kernel<<<gemmBlocks, 128, 0, stream>>>(LX_X, Wcxh, zrow, Cxh);
  // 4) gates Z, R
  gemm_gate_kernel<0><<<gemmBlocks, 128, 0, stream>>>(Cxz, wxz, Chz, wqz, bz,
                                                      nullptr, nullptr, zrow, Zbuf);
  gemm_gate_kernel<0><<<gemmBlocks, 128, 0, stream>>>(Cxr, wxr, Chr, wqr, br,
                                                      nullptr, nullptr, zrow, Rbuf);
  // 5) HR = H * R, propagate, conv
  mul_kernel<<<(int)(NC / 256), 256, 0, stream>>>(H, Rbuf, HR, (int)NC);
  prop_kernel<<<propGrid, 256, 0, stream>>>(HR, LX_HR, nrm, ei, lam);
  gemm_relu_kernel<<<gemmBlocks, 128, 0, stream>>>(LX_HR, Wchh, zrow, Chh);
  // 6) H_tilde + final blend -> out
  gemm_gate_kernel<1><<<gemmBlocks, 128, 0, stream>>>(Cxh, wxh, Chh, wqh, bh,
                                                      Zbuf, H, zrow, out);
}